// YOLOv1_LOSS_16063177687101
// MI455X (gfx1250) — compile-verified
//
#include <hip/hip_runtime.h>
#include <hip/hip_bf16.h>

#define GRID_NUM 7
#define CELL_PRE 30
#define CELL_TRUE 8
#define LAMBDA_COORD 5.0f
#define LAMBDA_NOOBJ 0.5f
#define EPSV 1e-12f
#define IMG 448.0f
#define GS 64.0f   // IMG / GRID_NUM

typedef __attribute__((ext_vector_type(2))) float f32x2;
typedef __attribute__((ext_vector_type(4))) float f32x4;
typedef __attribute__((ext_vector_type(8))) float v8f;

// Exact fp32 sum of all 32 lanes' `p`, returned in every lane.
// Uses V_WMMA_F32_16X16X4_F32 with A = per-lane partials (16x4 layout:
// lanes 0-15 hold K=0, lanes 16-31 hold K=2; second VGPR zero) and B = ones.
// D[m,n] = p(m) + p(m+16); lane L<16 holds rows 0..7 (its 8 acc VGPRs),
// lane L>=16 holds rows 8..15, so per-lane VGPR sum + xor-16 swizzle = total.
__device__ __forceinline__ float wave_sum32(float p) {
    f32x2 a; a.x = p;    a.y = 0.0f;
    f32x2 b; b.x = 1.0f; b.y = 1.0f;
    v8f c = {};
    v8f d = __builtin_amdgcn_wmma_f32_16x16x4_f32(
        /*neg_a=*/false, a, /*neg_b=*/false, b,
        /*c_mod=*/(short)0, c, /*reuse_a=*/false, /*reuse_b=*/false);
    float s = d[0] + d[1] + d[2] + d[3] + d[4] + d[5] + d[6] + d[7];
    // XOR-16 swap via ds_swizzle (group-of-32: xor=0x10, and=0x1f) -> 0x401F
    s += __int_as_float(__builtin_amdgcn_ds_swizzle(__float_as_int(s), 0x401F));
    return s;
}

__device__ __forceinline__ float block_sum256(float p, float* smem8) {
    float w = wave_sum32(p);
    const int lane = threadIdx.x & 31;
    const int wid  = threadIdx.x >> 5;
    if (lane == 0) smem8[wid] = w;
    __syncthreads();
    float t = 0.0f;
    if (threadIdx.x == 0) {
#pragma unroll
        for (int i = 0; i < 8; ++i) t += smem8[i];
    }
    return t;  // valid in thread 0 only
}

__global__ __launch_bounds__(256)
void yolo_loss_stage1(const float* __restrict__ yp,
                      const float* __restrict__ yt,
                      float* __restrict__ block_out,
                      int ncells) {
    const int cell = blockIdx.x * 256 + threadIdx.x;
    float loss = 0.0f;

    if (cell < ncells) {
        // y_pre: 30 contiguous floats, cell base is 8B-aligned -> 15x float2 (NT stream)
        const f32x2* pp = (const f32x2*)(yp + (size_t)cell * CELL_PRE);
        float p[CELL_PRE];
#pragma unroll
        for (int i = 0; i < 15; ++i) {
            f32x2 v = __builtin_nontemporal_load(pp + i);
            p[2 * i]     = v.x;
            p[2 * i + 1] = v.y;
        }
        // y_true: 8 floats, 32B-aligned -> 2x float4
        const f32x4* tp = (const f32x4*)(yt + (size_t)cell * CELL_TRUE);
        f32x4 t0 = __builtin_nontemporal_load(tp);
        f32x4 t1 = __builtin_nontemporal_load(tp + 1);
        const float tcls = t0.x;
        const float gx = t0.y, gy = t0.z, gw = t0.w, gh = t1.x;

        // IoU of both predicted boxes vs gt
        const float cx2 = gx * GS, cy2 = gy * GS;
        const float w2 = gw * IMG, h2 = gh * IMG;
        const float area2 = w2 * h2;
        auto iou = [&](float bx, float by, float bw, float bh) -> float {
            float cx1 = bx * GS, cy1 = by * GS;
            float w1 = bw * IMG, h1 = bh * IMG;
            float ix = fmaxf(fminf(cx1 + 0.5f * w1, cx2 + 0.5f * w2) -
                             fmaxf(cx1 - 0.5f * w1, cx2 - 0.5f * w2), 0.0f);
            float iy = fmaxf(fminf(cy1 + 0.5f * h1, cy2 + 0.5f * h2) -
                             fmaxf(cy1 - 0.5f * h1, cy2 - 0.5f * h2), 0.0f);
            float inter = ix * iy;
            return inter / (w1 * h1 + area2 - inter + EPSV);
        };
        const float iou0 = iou(p[0], p[1], p[2], p[3]);
        const float iou1 = iou(p[5], p[6], p[7], p[8]);
        const bool  c0 = iou0 > iou1;
        const float om = (tcls != 0.0f) ? 1.0f : 0.0f;

        // confidence loss
        const float conf_pre  = c0 ? p[4] : p[9];
        const float conf_true = c0 ? iou0 : iou1;
        const float dc = conf_pre - conf_true;
        loss  = om * dc * dc;
        loss += (1.0f - om) * LAMBDA_NOOBJ * (p[4] * p[4] + p[9] * p[9]);

        // coord + scale loss (chosen box)
        const float xpre = c0 ? p[0] : p[5];
        const float ypre = c0 ? p[1] : p[6];
        const float wpre = c0 ? p[2] : p[7];
        const float hpre = c0 ? p[3] : p[8];
        const float dx = xpre - gx, dy = ypre - gy;
        loss += LAMBDA_COORD * om * (dx * dx + dy * dy);
        const float dw = sqrtf(wpre + EPSV) - sqrtf(gw + EPSV);
        const float dh = sqrtf(hpre + EPSV) - sqrtf(gh + EPSV);
        loss += LAMBDA_COORD * om * (dw * dw + dh * dh);

        // class loss vs one_hot(class_id - 1)
        const int cid = (int)tcls - 1;
        float cls = 0.0f;
#pragma unroll
        for (int ci = 0; ci < 20; ++ci) {
            float d = p[10 + ci] - ((ci == cid) ? 1.0f : 0.0f);
            cls += d * d;
        }
        loss += om * cls;
    }

    __shared__ float smem[8];
    float bsum = block_sum256(loss, smem);   // EXEC all-1s here
    if (threadIdx.x == 0) block_out[blockIdx.x] = bsum;
}

__global__ __launch_bounds__(256)
void yolo_loss_stage2(const float* __restrict__ parts, int n,
                      float* __restrict__ out, float scale) {
    float acc = 0.0f;
    for (int i = threadIdx.x; i < n; i += 256) acc += parts[i];
    __shared__ float smem[8];
    float total = block_sum256(acc, smem);   // reconverged before WMMA
    if (threadIdx.x == 0) out[0] = total * scale;
}

extern "C" void kernel_launch(void* const* d_in, const int* in_sizes, int n_in,
                              void* d_out, int out_size, void* d_ws, size_t ws_size,
                              hipStream_t stream) {
    const float* yp = (const float*)d_in[0];   // (B,7,7,30) fp32
    const float* yt = (const float*)d_in[1];   // (B,7,7,8)  fp32
    const int ncells  = in_sizes[0] / CELL_PRE;          // B*49
    const int nblocks = (ncells + 255) / 256;            // 3136 for B=16384
    const int batch   = ncells / (GRID_NUM * GRID_NUM);  // B
    float* parts = (float*)d_ws;                         // nblocks floats of scratch

    yolo_loss_stage1<<<nblocks, 256, 0, stream>>>(yp, yt, parts, ncells);
    yolo_loss_stage2<<<1, 256, 0, stream>>>(parts, nblocks, (float*)d_out,
                                            1.0f / (float)batch);
}